// NodeAttentionLayer_50938312131111
// MI455X (gfx1250) — compile-verified
//
#include <hip/hip_runtime.h>
#include <math.h>

typedef float v2f __attribute__((ext_vector_type(2)));
typedef float v8f __attribute__((ext_vector_type(8)));

#define NEG_SLOPE 0.05f
#define D 64
#define FN 128
#define FE 64

__device__ __forceinline__ float leaky(float x) { return x >= 0.f ? x : NEG_SLOPE * x; }

// monotonic order-preserving float<->uint encoding for atomicMax on floats
__device__ __forceinline__ unsigned enc_f32(float f) {
    unsigned u = __float_as_uint(f);
    return (u & 0x80000000u) ? ~u : (u | 0x80000000u);
}
__device__ __forceinline__ float dec_f32(unsigned e) {
    return (e & 0x80000000u) ? __uint_as_float(e & 0x7FFFFFFFu) : __uint_as_float(~e);
}

// ---------------------------------------------------------------- init
__global__ void k_zero(float* __restrict__ p, long long n) {
    long long i = (long long)blockIdx.x * blockDim.x + threadIdx.x;
    long long stride = (long long)gridDim.x * blockDim.x;
    for (; i < n; i += stride) p[i] = 0.f;
}

// ---------------------------------------------------------------- h = x @ Wn^T  (WMMA f32 16x16x4)
// one wave (32 lanes) per 16-row tile; computes all 4 column tiles of D=64
__global__ void k_gemm_node(const float* __restrict__ x, const float* __restrict__ Wn,
                            float* __restrict__ h, int N) {
    const int lane = threadIdx.x;
    const int half = lane >> 4;    // selects K pair {0,1} vs {2,3}
    const int lm   = lane & 15;    // row (A) / col (B,C,D)
    const int row0 = blockIdx.x * 16;

    v8f acc[4];
    for (int t = 0; t < 4; ++t)
        for (int r = 0; r < 8; ++r) acc[t][r] = 0.f;

    const int rm = min(row0 + lm, N - 1);  // clamped load row (uniform-safe)
    for (int k = 0; k < FN; k += 4) {
        v2f a;
        const float* xr = x + (long long)rm * FN + k + 2 * half;
        a.x = xr[0];
        a.y = xr[1];
        for (int t = 0; t < 4; ++t) {
            v2f b;  // B[kk][n] = Wn[col=n][k+kk]  (B = Wn^T tile)
            const float* wr = Wn + (long long)(t * 16 + lm) * FN + k + 2 * half;
            b.x = wr[0];
            b.y = wr[1];
            acc[t] = __builtin_amdgcn_wmma_f32_16x16x4_f32(
                false, a, false, b, (short)0, acc[t], false, false);
        }
    }
    // C/D layout: VGPR r, lanes<16 -> M=r, lanes>=16 -> M=r+8, col = lm
    for (int t = 0; t < 4; ++t)
        for (int r = 0; r < 8; ++r) {
            int m = row0 + r + 8 * half;
            if (m < N) h[(long long)m * D + t * 16 + lm] = acc[t][r];
        }
}

// ---------------------------------------------------------------- g = edge_attr @ We^T  + fused ge = g @ a_edge[D:]
__global__ void k_gemm_edge(const float* __restrict__ ea, const float* __restrict__ We,
                            const float* __restrict__ a_edge,
                            float* __restrict__ g, float* __restrict__ ge, int E) {
    const int lane = threadIdx.x;
    const int half = lane >> 4;
    const int lm   = lane & 15;
    const int row0 = blockIdx.x * 16;

    v8f acc[4];
    for (int t = 0; t < 4; ++t)
        for (int r = 0; r < 8; ++r) acc[t][r] = 0.f;

    const int rm = min(row0 + lm, E - 1);
    for (int k = 0; k < FE; k += 4) {
        v2f a;
        const float* xr = ea + (long long)rm * FE + k + 2 * half;
        a.x = xr[0];
        a.y = xr[1];
        for (int t = 0; t < 4; ++t) {
            v2f b;
            const float* wr = We + (long long)(t * 16 + lm) * FE + k + 2 * half;
            b.x = wr[0];
            b.y = wr[1];
            acc[t] = __builtin_amdgcn_wmma_f32_16x16x4_f32(
                false, a, false, b, (short)0, acc[t], false, false);
        }
    }
    // store g + fused ge reduction (saves a full 205MB re-read of g)
    for (int r = 0; r < 8; ++r) {
        int m = row0 + r + 8 * half;
        float pv = 0.f;
        for (int t = 0; t < 4; ++t) {
            if (m < E) g[(long long)m * D + t * 16 + lm] = acc[t][r];
            pv += acc[t][r] * a_edge[D + t * 16 + lm];
        }
        // butterfly over the 16 lanes of each half (bits 0..3 stay within half)
        for (int off = 1; off < 16; off <<= 1) pv += __shfl_xor(pv, off, 32);
        if (lm == 0 && m < E) ge[m] = pv;
    }
}

// ---------------------------------------------------------------- hi/hj/he = h @ a vectors
__global__ void k_vec_node(const float* __restrict__ h, const float* __restrict__ a_node,
                           const float* __restrict__ a_edge,
                           float* __restrict__ hi, float* __restrict__ hj,
                           float* __restrict__ he, int N) {
    int n = blockIdx.x * blockDim.x + threadIdx.x;
    if (n >= N) return;
    float s1 = 0.f, s2 = 0.f, s3 = 0.f;
    const float* hr = h + (long long)n * D;
    for (int d = 0; d < D; ++d) {
        float v = hr[d];
        s1 += v * a_node[d];
        s2 += v * a_node[D + d];
        s3 += v * a_edge[d];
    }
    hi[n] = s1;
    hj[n] = s2;
    he[n] = s3;
}

// ---------------------------------------------------------------- segment max (node branch, E edges + N self loops)
__global__ void k_max_node(const long long* __restrict__ ei,
                           const float* __restrict__ hi, const float* __restrict__ hj,
                           unsigned* __restrict__ m_n, int E, int N) {
    int gid = blockIdx.x * blockDim.x + threadIdx.x;
    if (gid >= E + N) return;
    int i, j;
    if (gid < E) { i = (int)ei[gid]; j = (int)ei[(long long)E + gid]; }
    else         { i = j = gid - E; }
    float e = leaky(hi[i] + hj[j]);
    atomicMax(&m_n[i], enc_f32(e));
}

// ---------------------------------------------------------------- segment max (edge branch, 2E entries)
__global__ void k_max_edge(const long long* __restrict__ ei,
                           const float* __restrict__ he, const float* __restrict__ ge,
                           unsigned* __restrict__ m_e, int E) {
    int gid = blockIdx.x * blockDim.x + threadIdx.x;
    if (gid >= 2 * E) return;
    int k = (gid < E) ? gid : gid - E;
    int i = (gid < E) ? (int)ei[gid] : (int)ei[(long long)E + (gid - E)];
    float e = leaky(he[i] + ge[k]);
    atomicMax(&m_e[i], enc_f32(e));
}

// ---------------------------------------------------------------- scatter node branch: wave per entry, 2 feats/lane
__global__ void k_scatter_node(const long long* __restrict__ ei,
                               const float* __restrict__ hi, const float* __restrict__ hj,
                               const float* __restrict__ h,
                               const unsigned* __restrict__ m_n,
                               float* __restrict__ s_n, float* __restrict__ cnt_n,
                               float* __restrict__ wsum_n, int E, int N) {
    int t = blockIdx.x * blockDim.x + threadIdx.x;
    int entry = t >> 5;
    int lane  = t & 31;
    if (entry >= E + N) return;
    int i, j;
    if (entry < E) { i = (int)ei[entry]; j = (int)ei[(long long)E + entry]; }
    else           { i = j = entry - E; }
    float e  = leaky(hi[i] + hj[j]);
    float ex = __expf(e - dec_f32(m_n[i]));
    if (lane == 0) {
        atomicAdd(&s_n[i], ex);
        atomicAdd(&cnt_n[i], 1.f);
    }
    const float* hr = h + (long long)j * D + lane * 2;
    float* wp = wsum_n + (long long)i * D + lane * 2;
    atomicAdd(&wp[0], ex * hr[0]);
    atomicAdd(&wp[1], ex * hr[1]);
}

// ---------------------------------------------------------------- scatter edge branch
__global__ void k_scatter_edge(const long long* __restrict__ ei,
                               const float* __restrict__ he, const float* __restrict__ ge,
                               const float* __restrict__ g,
                               const unsigned* __restrict__ m_e,
                               float* __restrict__ s_e, float* __restrict__ cnt_e,
                               float* __restrict__ wsum_e, int E) {
    int t = blockIdx.x * blockDim.x + threadIdx.x;
    int entry = t >> 5;
    int lane  = t & 31;
    if (entry >= 2 * E) return;
    int k = (entry < E) ? entry : entry - E;
    int i = (entry < E) ? (int)ei[entry] : (int)ei[(long long)E + (entry - E)];
    float e  = leaky(he[i] + ge[k]);
    float ex = __expf(e - dec_f32(m_e[i]));
    if (lane == 0) {
        atomicAdd(&s_e[i], ex);
        atomicAdd(&cnt_e[i], 1.f);
    }
    const float* gr = g + (long long)k * D + lane * 2;
    float* wp = wsum_e + (long long)i * D + lane * 2;
    atomicAdd(&wp[0], ex * gr[0]);
    atomicAdd(&wp[1], ex * gr[1]);
}

// ---------------------------------------------------------------- finalize
__global__ void k_finalize(const float* __restrict__ s_n, const float* __restrict__ cnt_n,
                           const float* __restrict__ wsum_n,
                           const float* __restrict__ s_e, const float* __restrict__ cnt_e,
                           const float* __restrict__ wsum_e,
                           float* __restrict__ out, int N) {
    int gid = blockIdx.x * blockDim.x + threadIdx.x;
    if (gid >= N * D) return;
    int n = gid >> 6;
    int d = gid & 63;
    float sn = s_n[n], cn = fmaxf(cnt_n[n], 1.f);
    float vn = (sn > 0.f) ? wsum_n[(long long)n * D + d] / (sn * cn) : 0.f;
    float se = s_e[n], ce = fmaxf(cnt_e[n], 1.f);
    float ve = (se > 0.f) ? wsum_e[(long long)n * D + d] / (se * ce) : 0.f;
    out[(long long)n * (2 * D) + d]     = leaky(vn);
    out[(long long)n * (2 * D) + D + d] = leaky(ve);
}

extern "C" void kernel_launch(void* const* d_in, const int* in_sizes, int n_in,
                              void* d_out, int out_size, void* d_ws, size_t ws_size,
                              hipStream_t stream) {
    const float*     x       = (const float*)d_in[0];
    const float*     ea      = (const float*)d_in[1];
    const long long* ei      = (const long long*)d_in[2];  // jnp.int64
    const float*     Wn      = (const float*)d_in[3];
    const float*     We      = (const float*)d_in[4];
    const float*     a_node  = (const float*)d_in[5];
    const float*     a_edge  = (const float*)d_in[6];
    float* out = (float*)d_out;

    const int N = in_sizes[0] / FN;
    const int E = in_sizes[2] / 2;

    float* ws = (float*)d_ws;
    size_t off = 0;
    float* h  = ws + off; off += (size_t)N * D;
    float* g  = ws + off; off += (size_t)E * D;
    float* hi = ws + off; off += N;
    float* hj = ws + off; off += N;
    float* he = ws + off; off += N;
    float* ge = ws + off; off += E;
    float*    statsBase = ws + off;
    unsigned* m_n    = (unsigned*)(statsBase);
    float*    s_n    = statsBase + N;
    float*    cnt_n  = statsBase + 2 * (size_t)N;
    float*    wsum_n = statsBase + 3 * (size_t)N;
    unsigned* m_e    = (unsigned*)(statsBase + 3 * (size_t)N + (size_t)N * D);
    float*    s_e    = statsBase + 4 * (size_t)N + (size_t)N * D;
    float*    cnt_e  = statsBase + 5 * (size_t)N + (size_t)N * D;
    float*    wsum_e = statsBase + 6 * (size_t)N + (size_t)N * D;
    const long long statsLen = (long long)N * (2 * D + 6);

    // zero accumulators / max-encodings (enc 0 acts as -inf identity)
    k_zero<<<2048, 256, 0, stream>>>(statsBase, statsLen);

    // dense phase (WMMA fp32)
    k_gemm_node<<<(N + 15) / 16, 32, 0, stream>>>(x, Wn, h, N);
    k_vec_node<<<(N + 255) / 256, 256, 0, stream>>>(h, a_node, a_edge, hi, hj, he, N);
    k_gemm_edge<<<(E + 15) / 16, 32, 0, stream>>>(ea, We, a_edge, g, ge, E);

    // segment max
    k_max_node<<<(E + N + 255) / 256, 256, 0, stream>>>(ei, hi, hj, m_n, E, N);
    k_max_edge<<<(2 * E + 255) / 256, 256, 0, stream>>>(ei, he, ge, m_e, E);

    // exp + weighted scatter (wave per entry)
    {
        long long th = (long long)(E + N) * 32;
        k_scatter_node<<<(unsigned)((th + 255) / 256), 256, 0, stream>>>(
            ei, hi, hj, h, m_n, s_n, cnt_n, wsum_n, E, N);
    }
    {
        long long th = (long long)(2 * E) * 32;
        k_scatter_edge<<<(unsigned)((th + 255) / 256), 256, 0, stream>>>(
            ei, he, ge, g, m_e, s_e, cnt_e, wsum_e, E);
    }

    k_finalize<<<(N * D + 255) / 256, 256, 0, stream>>>(
        s_n, cnt_n, wsum_n, s_e, cnt_e, wsum_e, out, N);
}